// SAGE_43782896615725
// MI455X (gfx1250) — compile-verified
//
#include <hip/hip_runtime.h>

// ---------------------------------------------------------------------------
// GraphSAGE (3 layers) + mean readout for MI455X (gfx1250, wave32, WMMA).
//
// GEMMs use V_WMMA_F32_16X16X4_F32 fed from an LDS-staged weight panel
// (async global->LDS copy via global_load_async_to_lds_b64, ASYNCcnt path).
// Edge aggregation (the real bottleneck, ~0.5 GB of L2-resident traffic)
// uses non-returning global_atomic_add_f32.
// ---------------------------------------------------------------------------

#define N_NODES   10000
#define N_EDGES   320000
#define N_GRAPHS  16
#define ROWBLKS   (N_NODES / 16)   // 625, exact

typedef __attribute__((ext_vector_type(2))) float v2f;
typedef __attribute__((ext_vector_type(8))) float v8f;
typedef __attribute__((ext_vector_type(2))) int   v2i;

typedef __attribute__((address_space(1))) v2i* gas_v2i_ptr;
typedef __attribute__((address_space(3))) v2i* las_v2i_ptr;

__global__ void fill_zero(float* __restrict__ p, int n) {
  int i = blockIdx.x * blockDim.x + threadIdx.x;
  if (i < n) p[i] = 0.0f;
}

// Pack [W_self | W_neigh] (each K x 64) into the WMMA B-operand (4x16 f32)
// layout: Wp[(kstep*8 + ct)*32 + lane] = { W[k0][col], W[k0+1][col] }
// with hi = lane>>4, k0 = kstep*4 + 2*hi, col = ct*16 + (lane&15).
// (B mirrors the ISA A-layout: VGPR v, lane-half hi -> K = v + 2*hi.)
__global__ void pack_w(const float* __restrict__ Ws, const float* __restrict__ Wn,
                       float* __restrict__ Wp, int K) {
  int t = blockIdx.x * blockDim.x + threadIdx.x;   // over (K/4)*8*32
  int total = (K >> 2) * 8 * 32;
  if (t >= total) return;
  int lane  = t & 31;
  int ct    = (t >> 5) & 7;
  int kstep = t >> 8;
  int hi = lane >> 4, n = lane & 15;
  int k0  = kstep * 4 + 2 * hi;
  int col = ct * 16 + n;
  float w0, w1;
  if (col < 64) { w0 = Ws[k0 * 64 + col];        w1 = Ws[(k0 + 1) * 64 + col]; }
  else          { w0 = Wn[k0 * 64 + col - 64];   w1 = Wn[(k0 + 1) * 64 + col - 64]; }
  ((float2*)Wp)[t] = make_float2(w0, w1);
}

// P[N x 128] = A[N x K] @ [W_self | W_neigh][K x 128] via WMMA f32 16x16x4.
// blockIdx.y selects the 64-column half; each wave owns one 16-row block and
// 4 column tiles (32 accumulator VGPRs -> no spills). The packed B panel for
// this half is staged in LDS once per block and read with ds_load_b64.
template <int KSTEPS>
__global__ void gemm_wmma(const float* __restrict__ A, const float* __restrict__ Wp,
                          float* __restrict__ P) {
  constexpr int K = KSTEPS * 4;
  __shared__ float2 Bs[KSTEPS * 4 * 32];       // this half's packed B panel
  const int t    = threadIdx.x;                 // 256 threads = 8 waves
  const int half = blockIdx.y;                  // 0: cols 0..63, 1: 64..127

  // ---- cooperative stage of packed B into LDS ----
  const int n2 = KSTEPS * 4 * 32;               // float2 elements
#if __has_builtin(__builtin_amdgcn_global_load_async_to_lds_b64)
  for (int i = t; i < n2; i += 256) {
    int lane = i & 31, c = (i >> 5) & 3, ks = i >> 7;
    const float2* gsrc = ((const float2*)Wp) + ((ks * 8 + half * 4 + c) * 32 + lane);
    __builtin_amdgcn_global_load_async_to_lds_b64(
        (gas_v2i_ptr)gsrc, (las_v2i_ptr)&Bs[i], 0, 0);
  }
  asm volatile("s_wait_asynccnt 0" ::: "memory");
#else
  for (int i = t; i < n2; i += 256) {
    int lane = i & 31, c = (i >> 5) & 3, ks = i >> 7;
    Bs[i] = ((const float2*)Wp)[(ks * 8 + half * 4 + c) * 32 + lane];
  }
#endif
  __syncthreads();

  const int wave = t >> 5;
  const int lane = t & 31;
  const int rb   = blockIdx.x * 8 + wave;
  const int hi = lane >> 4, m = lane & 15, n = m;

  if (rb < ROWBLKS) {                          // wave-uniform: EXEC stays full
    const int row0 = rb * 16;
    v8f acc[4] = {};
    for (int ks = 0; ks < KSTEPS; ++ks) {
      v2f a = *(const v2f*)&A[(long)(row0 + m) * K + ks * 4 + 2 * hi];
#pragma unroll
      for (int c = 0; c < 4; ++c) {
        v2f b = *(const v2f*)&Bs[(ks * 4 + c) * 32 + lane];
        acc[c] = __builtin_amdgcn_wmma_f32_16x16x4_f32(
            false, a, false, b, (short)0, acc[c], false, false);
      }
    }
    // C/D layout: VGPR v, lane-half hi -> M = v + 8*hi; N = lane&15.
    const int orow = row0 + 8 * hi;
    const int col0 = half * 64;
#pragma unroll
    for (int c = 0; c < 4; ++c)
#pragma unroll
      for (int v = 0; v < 8; ++v)
        P[(long)(orow + v) * 128 + col0 + c * 16 + n] = acc[c][v];
  }
}

__global__ void degree_kernel(const int* __restrict__ dst, float* __restrict__ deg) {
  int e = blockIdx.x * blockDim.x + threadIdx.x;
  if (e < N_EDGES) atomicAdd(&deg[dst[e]], 1.0f);
}

// agg[dst[e]][c] += P[src[e]][64 + c]  (projected-neighbor columns of P)
// 16 threads/edge, float4 each: fully coalesced 256B gather + scatter.
__global__ void edge_aggregate(const float* __restrict__ P, const int* __restrict__ src,
                               const int* __restrict__ dst, float* __restrict__ agg) {
  long t = (long)blockIdx.x * blockDim.x + threadIdx.x;
  int e = (int)(t >> 4);
  if (e >= N_EDGES) return;
  int c4 = ((int)t & 15) * 4;
  int s = src[e], d = dst[e];
  float4 v = *(const float4*)&P[(long)s * 128 + 64 + c4];
  float* o = &agg[(long)d * 64 + c4];
  atomicAdd(o + 0, v.x);
  atomicAdd(o + 1, v.y);
  atomicAdd(o + 2, v.z);
  atomicAdd(o + 3, v.w);
}

// out[i][c] = act( P_self[i][c] + agg[i][c]/max(deg[i],1) + b[c] )
__global__ void combine(const float* __restrict__ P, const float* __restrict__ agg,
                        const float* __restrict__ deg, const float* __restrict__ bias,
                        float* __restrict__ out, int relu) {
  int t = blockIdx.x * blockDim.x + threadIdx.x;
  int node = t >> 4;
  if (node >= N_NODES) return;
  int c4 = (t & 15) * 4;
  float invd = 1.0f / fmaxf(deg[node], 1.0f);
  float4 s = *(const float4*)&P[(long)node * 128 + c4];
  float4 a = *(const float4*)&agg[(long)node * 64 + c4];
  float4 b = *(const float4*)&bias[c4];
  float4 r;
  r.x = s.x + a.x * invd + b.x;
  r.y = s.y + a.y * invd + b.y;
  r.z = s.z + a.z * invd + b.z;
  r.w = s.w + a.w * invd + b.w;
  if (relu) {
    r.x = fmaxf(r.x, 0.0f); r.y = fmaxf(r.y, 0.0f);
    r.z = fmaxf(r.z, 0.0f); r.w = fmaxf(r.w, 0.0f);
  }
  *(float4*)&out[(long)node * 64 + c4] = r;
}

// Per-block LDS partial segment sums, then a handful of global atomics.
__global__ void readout_partial(const float* __restrict__ h, const int* __restrict__ gid,
                                float* __restrict__ gsum, float* __restrict__ gcnt) {
  __shared__ float lsum[N_GRAPHS * 64];
  __shared__ float lcnt[N_GRAPHS];
  int t = threadIdx.x;                         // 256 threads
  for (int i = t; i < N_GRAPHS * 64; i += 256) lsum[i] = 0.0f;
  if (t < N_GRAPHS) lcnt[t] = 0.0f;
  __syncthreads();
  int base = blockIdx.x * 256;                 // 256 nodes per block
  for (int i = t; i < 256 * 64; i += 256) {
    int node = base + (i >> 6);
    if (node < N_NODES) {
      int c = i & 63;
      int g = gid[node];
      atomicAdd(&lsum[g * 64 + c], h[(long)node * 64 + c]);
      if (c == 0) atomicAdd(&lcnt[g], 1.0f);
    }
  }
  __syncthreads();
  for (int i = t; i < N_GRAPHS * 64; i += 256) atomicAdd(&gsum[i], lsum[i]);
  if (t < N_GRAPHS) atomicAdd(&gcnt[t], lcnt[t]);
}

// out_feat = gsum/max(gcnt,1);  out = out_feat @ W_cls + b_cls
__global__ void readout_finalize(const float* __restrict__ gsum, const float* __restrict__ gcnt,
                                 const float* __restrict__ Wc, const float* __restrict__ bc,
                                 float* __restrict__ out) {
  __shared__ float of[N_GRAPHS * 64];
  int t = threadIdx.x;                          // 1024 threads, 1 block
  if (t < N_GRAPHS * 64) {
    int g = t >> 6;
    float v = gsum[t] / fmaxf(gcnt[g], 1.0f);
    of[t] = v;
    out[32 + t] = v;                            // out_feat region
  }
  __syncthreads();
  if (t < N_GRAPHS * 2) {
    int g = t >> 1, k = t & 1;
    float acc = bc[k];
    for (int c = 0; c < 64; ++c) acc += of[g * 64 + c] * Wc[c * 2 + k];
    out[g * 2 + k] = acc;                       // out region
  }
}

extern "C" void kernel_launch(void* const* d_in, const int* in_sizes, int n_in,
                              void* d_out, int out_size, void* d_ws, size_t ws_size,
                              hipStream_t stream) {
  const float* feat = (const float*)d_in[0];
  const int*   src  = (const int*)d_in[1];
  const int*   dst  = (const int*)d_in[2];
  const int*   gid  = (const int*)d_in[3];
  const float* Ws1 = (const float*)d_in[4];
  const float* Wn1 = (const float*)d_in[5];
  const float* b1  = (const float*)d_in[6];
  const float* Ws2 = (const float*)d_in[7];
  const float* Wn2 = (const float*)d_in[8];
  const float* b2  = (const float*)d_in[9];
  const float* Ws3 = (const float*)d_in[10];
  const float* Wn3 = (const float*)d_in[11];
  const float* b3  = (const float*)d_in[12];
  const float* Wc  = (const float*)d_in[13];
  const float* bc  = (const float*)d_in[14];

  float* out = (float*)d_out;                    // [32 | 1024 | 640000]
  float* h3  = out + 32 + N_GRAPHS * 64;         // h written straight to d_out

  // workspace carve-out (floats)
  float* ws   = (float*)d_ws;
  float* P    = ws;                  // 10000*128 = 1,280,000
  float* agg  = P    + 1280000;      // 10000*64  =   640,000
  float* deg  = agg  + 640000;       // 10,000 (padded to 10,240)
  float* hbuf = deg  + 10240;        // 10000*64  =   640,000
  float* Wp   = hbuf + 640000;       // (256/4)*8*32*2 = 32,768 max
  float* gsum = Wp   + 32768;        // 1024
  float* gcnt = gsum + 1024;         // 16   (contiguous with gsum)

  const int B = 256;
  dim3 blk(B);
  dim3 gemm_grid((ROWBLKS + 7) / 8, 2);                    // 8 waves x 2 halves
  int edge_grid = (int)(((long)N_EDGES * 16 + B - 1) / B); // 16 thr/edge
  int comb_grid = (N_NODES * 16 + B - 1) / B;

  // in-degree (dst only, shared by all layers)
  fill_zero<<<(N_NODES + B - 1) / B, blk, 0, stream>>>(deg, N_NODES);
  degree_kernel<<<(N_EDGES + B - 1) / B, blk, 0, stream>>>(dst, deg);

  // ---- layer 1 (K = 256, relu) ----
  pack_w<<<((256 / 4) * 8 * 32 + B - 1) / B, blk, 0, stream>>>(Ws1, Wn1, Wp, 256);
  gemm_wmma<64><<<gemm_grid, blk, 0, stream>>>(feat, Wp, P);
  fill_zero<<<(640000 + B - 1) / B, blk, 0, stream>>>(agg, 640000);
  edge_aggregate<<<edge_grid, blk, 0, stream>>>(P, src, dst, agg);
  combine<<<comb_grid, blk, 0, stream>>>(P, agg, deg, b1, hbuf, 1);

  // ---- layer 2 (K = 64, relu) ----
  pack_w<<<((64 / 4) * 8 * 32 + B - 1) / B, blk, 0, stream>>>(Ws2, Wn2, Wp, 64);
  gemm_wmma<16><<<gemm_grid, blk, 0, stream>>>(hbuf, Wp, P);
  fill_zero<<<(640000 + B - 1) / B, blk, 0, stream>>>(agg, 640000);
  edge_aggregate<<<edge_grid, blk, 0, stream>>>(P, src, dst, agg);
  combine<<<comb_grid, blk, 0, stream>>>(P, agg, deg, b2, hbuf, 1);

  // ---- layer 3 (K = 64, no activation) -> h3 in d_out ----
  pack_w<<<((64 / 4) * 8 * 32 + B - 1) / B, blk, 0, stream>>>(Ws3, Wn3, Wp, 64);
  gemm_wmma<16><<<gemm_grid, blk, 0, stream>>>(hbuf, Wp, P);
  fill_zero<<<(640000 + B - 1) / B, blk, 0, stream>>>(agg, 640000);
  edge_aggregate<<<edge_grid, blk, 0, stream>>>(P, src, dst, agg);
  combine<<<comb_grid, blk, 0, stream>>>(P, agg, deg, b3, h3, 0);

  // ---- readout ----
  fill_zero<<<(1040 + B - 1) / B, blk, 0, stream>>>(gsum, 1040); // gsum + gcnt
  readout_partial<<<(N_NODES + 255) / 256, blk, 0, stream>>>(h3, gid, gsum, gcnt);
  readout_finalize<<<1, 1024, 0, stream>>>(gsum, gcnt, Wc, bc, out);
}